// MultiLabelBIOULDecoder_60722247631332
// MI455X (gfx1250) — compile-verified
//
#include <hip/hip_runtime.h>
#include <hip/hip_bf16.h>

// ---------------------------------------------------------------------------
// Constrained linear-chain CRF loss (BIOUL) on MI455X / gfx1250.
//
// alpha'[j] = logsumexp_i(alpha[i] + trans[i,j]) + e[j]
//           = m + log( sum_i exp(alpha[i]-m) * E[i,j] ) + e[j],  E = exp(trans_c)
// -> T=256 recursion of tiny f32 GEMMs on a FIXED 41x41 matrix, executed with
//    V_WMMA_F32_16X16X4_F32 (full-f32 WMMA, no precision loss).
//
//  * All 33 B-matrix v2f fragments of E are held in REGISTERS for the whole
//    time loop (66 VGPRs; only 512 waves exist chip-wide, so trading
//    occupancy for zero per-step B traffic is free).
//  * Per-step emissions/target are double-buffered into LDS with
//    global_load_async_to_lds_b32, issued one step ahead and consumed after
//    s_wait_asynccnt — copies fully overlap the WMMA work (ASYNCcnt path).
// ---------------------------------------------------------------------------

#define NTAGS       41
#define NPAD        48
#define KCHUNKS     11      // ceil(41/4)
#define NTILES      3       // ceil(41/16)
#define T_LEN       256
#define BATCH       4096
#define IMPOSSIBLE  (-1.0e7f)
#define PAD_NEG     (-1.0e30f)
#define WAVES_PER_BLOCK 4
#define E_PACK_FLOATS   (KCHUNKS * NTILES * 32 * 2)   // 2112

typedef float v2f __attribute__((ext_vector_type(2)));
typedef float v8f __attribute__((ext_vector_type(8)));

#define AS1 __attribute__((address_space(1)))
#define AS3 __attribute__((address_space(3)))

// Within-wave LDS handoff: DS ops from one wave complete in order at the LDS,
// so a dscnt wait + compiler memory fence is sufficient (no s_barrier needed).
#define LDS_FENCE() asm volatile("s_wait_dscnt 0" ::: "memory")

// ---- async copy global->LDS (ASYNCcnt-tracked), per-lane 4 bytes ----------
__device__ __forceinline__ void async_cp_b32(const void* g, void* l) {
#if __has_builtin(__builtin_amdgcn_global_load_async_to_lds_b32)
    __builtin_amdgcn_global_load_async_to_lds_b32(
        (AS1 int*)const_cast<void*>(g), (AS3 int*)l, 0, 0);
#else
    unsigned lds_off = (unsigned)(size_t)(AS3 void*)l;
    unsigned long long ga = (unsigned long long)(size_t)g;
    asm volatile("global_load_async_to_lds_b32 %0, %1, off"
                 :: "v"(lds_off), "v"(ga) : "memory");
#endif
}

#if __has_builtin(__builtin_amdgcn_s_wait_asynccnt)
#define ASYNC_WAIT(n) __builtin_amdgcn_s_wait_asynccnt(n)
#else
#define ASYNC_WAIT(n) asm volatile("s_wait_asynccnt " #n ::: "memory")
#endif

// ---------------------------------------------------------------------------
// Prep kernel: E = exp(constrained transitions), pre-packed in the WMMA
// B-fragment lane layout; constrained start/end vectors (padded to 48).
//
// B-fragment layout (32-bit B 4x16, mirroring documented A 16x4 layout):
//   lane 0-15 : v0 -> K=0, v1 -> K=1 ; lane 16-31 : v0 -> K=2, v1 -> K=3
//   column N = tile*16 + (lane & 15);  index ((kc*NTILES+nt)*32+lane)*2+v
// ---------------------------------------------------------------------------
__global__ void crf_prep_kernel(const float* __restrict__ trans,
                                const float* __restrict__ startT,
                                const float* __restrict__ endT,
                                const unsigned char* __restrict__ ft,
                                const unsigned char* __restrict__ sft,
                                const unsigned char* __restrict__ eft,
                                float* __restrict__ wsE,
                                float* __restrict__ wsStart,
                                float* __restrict__ wsEnd) {
    for (int idx = threadIdx.x; idx < E_PACK_FLOATS; idx += blockDim.x) {
        int frag = idx >> 6;          // kc*3 + nt
        int rem  = idx & 63;
        int lane = rem >> 1;
        int v    = rem & 1;
        int kc   = frag / NTILES;
        int nt   = frag % NTILES;
        int k    = kc * 4 + ((lane < 16) ? 0 : 2) + v;
        int n    = nt * 16 + (lane & 15);
        float val = 0.0f;
        if (k < NTAGS && n < NTAGS && !ft[k * NTAGS + n])
            val = __expf(trans[k * NTAGS + n]);
        wsE[idx] = val;
    }
    if (threadIdx.x < NPAD) {
        int j = threadIdx.x;
        wsStart[j] = (j < NTAGS) ? (sft[j] ? IMPOSSIBLE : startT[j]) : IMPOSSIBLE;
        wsEnd[j]   = (j < NTAGS) ? (eft[j] ? IMPOSSIBLE : endT[j])   : IMPOSSIBLE;
    }
}

// ---------------------------------------------------------------------------
// Main kernel: 128 threads = 4 waves; each wave owns one 16-row WMMA tile
// (8 batches x {supervised, unsupervised}).  Grid = 4096/32 = 128 blocks.
// ---------------------------------------------------------------------------
__global__ __launch_bounds__(32 * WAVES_PER_BLOCK, 1)
void crf_forward_kernel(const float* __restrict__ emis,
                        const unsigned char* __restrict__ mask,
                        const unsigned char* __restrict__ target,
                        const float* __restrict__ wsE,
                        const float* __restrict__ wsStart,
                        const float* __restrict__ wsEnd,
                        float* __restrict__ out) {
    __shared__ __align__(16) float sAlpha[WAVES_PER_BLOCK][16][NPAD];      // 12 KB
    __shared__ __align__(16) float sEmis[WAVES_PER_BLOCK][2][8][44];       // 11 KB
    __shared__ __align__(16) unsigned char sTgt[WAVES_PER_BLOCK][2][8][44];// 2.8 KB
    __shared__ float sM[WAVES_PER_BLOCK][16];
    __shared__ float sZ[WAVES_PER_BLOCK][16];
    __shared__ int   sLen[WAVES_PER_BLOCK][8];

    const int tid  = threadIdx.x;
    const int w    = tid >> 5;
    const int lane = tid & 31;
    const int batch0 = (blockIdx.x * WAVES_PER_BLOCK + w) * 8;

    // ---- hoist all 33 B fragments of E into registers for the whole loop --
    v2f Bf[KCHUNKS][NTILES];
    #pragma unroll
    for (int kc = 0; kc < KCHUNKS; ++kc)
        #pragma unroll
        for (int nt = 0; nt < NTILES; ++nt)
            Bf[kc][nt] = *(const v2f*)&wsE[((kc * NTILES + nt) * 32 + lane) * 2];

    // ---- per-batch base pointers -------------------------------------------
    const float*         ebase[8];
    const unsigned char* tbase[8];
    #pragma unroll
    for (int bi = 0; bi < 8; ++bi) {
        int b = batch0 + bi;
        ebase[bi] = emis   + (size_t)b * T_LEN * NTAGS;
        tbase[bi] = target + (size_t)b * T_LEN * NTAGS;   // dword-aligned (T*41*b)
    }

    // ---- sequence lengths (mask is left-aligned, 1 byte per bool) ----------
    if (lane < 8) {
        int b = batch0 + lane;
        const unsigned int* m32 = (const unsigned int*)(mask + (size_t)b * T_LEN);
        int len = 0;
        #pragma unroll 8
        for (int i = 0; i < T_LEN / 4; ++i) {
            unsigned int v = m32[i];
            len += (v & 1) + ((v >> 8) & 1) + ((v >> 16) & 1) + ((v >> 24) & 1);
        }
        sLen[w][lane] = len;
        sZ[w][2 * lane]     = 0.0f;
        sZ[w][2 * lane + 1] = 0.0f;
    }

    // ---- async stage of emissions+target for time tt into buffer bb --------
    // 8 batches x (41 f32 + 11 dwords of target bytes) = 24 async issues.
    auto stage = [&](int tt, int bb) {
        size_t eoff = (size_t)tt * NTAGS;                   // dwords
        size_t toff = ((size_t)tt * NTAGS) & ~(size_t)3;    // bytes, aligned down
        #pragma unroll
        for (int bi = 0; bi < 8; ++bi) {
            const float* gb = ebase[bi] + eoff;
            float*       lb = &sEmis[w][bb][bi][0];
            async_cp_b32(gb + lane, lb + lane);                       // dwords 0..31
            if (lane < NTAGS - 32)
                async_cp_b32(gb + 32 + lane, lb + 32 + lane);         // dwords 32..40
            const unsigned char* gt = tbase[bi] + toff;               // 4B aligned
            unsigned char*       lt = &sTgt[w][bb][bi][0];
            if (lane < 11)                                            // 44 bytes
                async_cp_b32((const float*)gt + lane, (float*)lt + lane);
        }
    };

    // ---- init alpha at t=0: alpha0 = e0 (+target mask for sup) + start -----
    {
        int r = lane & 15;
        int cbase = (lane < 16) ? 0 : 24;
        int b = batch0 + (r >> 1);
        bool sup = ((r & 1) == 0);
        for (int j = cbase; j < cbase + 24; ++j) {
            float val = PAD_NEG;
            if (j < NTAGS) {
                size_t eidx = ((size_t)b * T_LEN + 0) * NTAGS + j;
                float e = emis[eidx];
                if (sup && !target[eidx]) e = IMPOSSIBLE;
                val = e + wsStart[j];
            }
            sAlpha[w][r][j] = val;
        }
    }
    stage(1, 1);          // prefetch t=1 into buffer 1
    LDS_FENCE();

    const int kofs  = (lane < 16) ? 0 : 2;
    const int rA    = lane & 15;            // this lane's A-matrix row
    const int rhalf = (lane >= 16) ? 8 : 0; // C/D row offset for this half
    const int cmod  = lane & 15;            // column within N-tile

    // ---------------------------- time recursion ---------------------------
    for (int t = 1; t < T_LEN; ++t) {
        const int buf = t & 1;
        if (t < T_LEN - 1) stage(t + 1, buf ^ 1);   // overlap copy with compute

        // Row max (lanes 16-31 redundantly cover rows 0-15; dup store is benign).
        const float4* rowp = (const float4*)sAlpha[w][rA];
        float mmax = PAD_NEG;
        #pragma unroll
        for (int i = 0; i < NPAD / 4; ++i) {
            float4 vv = rowp[i];
            mmax = fmaxf(mmax, fmaxf(fmaxf(vv.x, vv.y), fmaxf(vv.z, vv.w)));
        }
        sM[w][rA] = mmax;
        LDS_FENCE();

        // u = exp(alpha - m) @ E  via 11 K-chunks x 3 N-tiles of f32 WMMA.
        v8f acc0 = {0,0,0,0,0,0,0,0};
        v8f acc1 = {0,0,0,0,0,0,0,0};
        v8f acc2 = {0,0,0,0,0,0,0,0};
        #pragma unroll
        for (int kc = 0; kc < KCHUNKS; ++kc) {
            int k0 = kc * 4 + kofs;
            v2f af;
            af.x = __expf(sAlpha[w][rA][k0]     - mmax);
            af.y = __expf(sAlpha[w][rA][k0 + 1] - mmax);
            acc0 = __builtin_amdgcn_wmma_f32_16x16x4_f32(false, af, false, Bf[kc][0],
                                                         (short)0, acc0, false, false);
            acc1 = __builtin_amdgcn_wmma_f32_16x16x4_f32(false, af, false, Bf[kc][1],
                                                         (short)0, acc1, false, false);
            acc2 = __builtin_amdgcn_wmma_f32_16x16x4_f32(false, af, false, Bf[kc][2],
                                                         (short)0, acc2, false, false);
        }
        LDS_FENCE();   // all alpha reads retired before the overwrite below

        float mrow[8];
        #pragma unroll
        for (int k = 0; k < 8; ++k) mrow[k] = sM[w][k + rhalf];

        // This step's staged emissions/targets are complete once the 24 copies
        // issued THIS step (for t+1) are the only outstanding ones.
        if (t < T_LEN - 1) ASYNC_WAIT(24); else ASYNC_WAIT(0);

        const int mis = (t * NTAGS) & 3;   // target byte misalignment (uniform)

        // alpha_new = m + log(u) + e_t   (sup rows: e masked by target)
        #pragma unroll
        for (int nt = 0; nt < NTILES; ++nt) {
            int col = nt * 16 + cmod;
            bool colok = (col < NTAGS);
            v8f acc = (nt == 0) ? acc0 : (nt == 1) ? acc1 : acc2;
            #pragma unroll
            for (int k = 0; k < 8; ++k) {
                int row = k + rhalf;
                float nv = PAD_NEG;
                if (colok) {
                    int bi = row >> 1;
                    float e = sEmis[w][buf][bi][col];
                    if (!(row & 1) && !sTgt[w][buf][bi][mis + col]) e = IMPOSSIBLE;
                    nv = mrow[k] + __logf(acc[k]) + e;
                    nv = fmaxf(nv, PAD_NEG);     // stop -inf/NaN cascades
                }
                sAlpha[w][row][col] = nv;
            }
        }
        LDS_FENCE();

        // Capture z = logsumexp(alpha + end) at the last valid token.
        if (lane < 16) {
            int row = lane;
            int b = row >> 1;
            if (t == sLen[w][b] - 1) {
                float m2 = PAD_NEG;
                float av[NTAGS];
                for (int j = 0; j < NTAGS; ++j) {
                    av[j] = sAlpha[w][row][j] + wsEnd[j];
                    m2 = fmaxf(m2, av[j]);
                }
                float s = 0.0f;
                for (int j = 0; j < NTAGS; ++j) s += __expf(av[j] - m2);
                sZ[w][row] = m2 + __logf(s);
            }
        }
    }
    LDS_FENCE();

    // loss[b] = -(sup_z - unsup_z) = z_unsup - z_sup
    if (lane < 8)
        out[batch0 + lane] = sZ[w][2 * lane + 1] - sZ[w][2 * lane];
}

// ---------------------------------------------------------------------------
extern "C" void kernel_launch(void* const* d_in, const int* in_sizes, int n_in,
                              void* d_out, int out_size, void* d_ws, size_t ws_size,
                              hipStream_t stream) {
    const float*         emissions = (const float*)d_in[0];
    const unsigned char* mask      = (const unsigned char*)d_in[1];
    const unsigned char* target    = (const unsigned char*)d_in[2];
    const float*         trans     = (const float*)d_in[3];
    const float*         startT    = (const float*)d_in[4];
    const float*         endT      = (const float*)d_in[5];
    const unsigned char* ft        = (const unsigned char*)d_in[6];
    const unsigned char* sft       = (const unsigned char*)d_in[7];
    const unsigned char* eft       = (const unsigned char*)d_in[8];
    float* out = (float*)d_out;

    float* wsf     = (float*)d_ws;
    float* wsE     = wsf;                    // 2112 floats
    float* wsStart = wsf + E_PACK_FLOATS;    // 48 floats
    float* wsEnd   = wsStart + NPAD;         // 48 floats

    crf_prep_kernel<<<1, 256, 0, stream>>>(trans, startT, endT, ft, sft, eft,
                                           wsE, wsStart, wsEnd);

    int blocks = BATCH / (8 * WAVES_PER_BLOCK);   // 128
    crf_forward_kernel<<<blocks, 32 * WAVES_PER_BLOCK, 0, stream>>>(
        emissions, mask, target, wsE, wsStart, wsEnd, out);
}